// GCN2Model_17635135718116
// MI455X (gfx1250) — compile-verified
//
#include <hip/hip_runtime.h>
#include <math.h>

#define HID 48
#define NTILES_N 3   // 48 / 16
#define KSTEPS 12    // 48 / 4
#define WPB 4        // waves per block in the WMMA kernel
#define WP 50        // padded LDS pitch for transposed W (bank-conflict-free b64 reads)
#define TP 50        // padded LDS pitch for T tile

typedef __attribute__((ext_vector_type(2))) float v2f;
typedef __attribute__((ext_vector_type(8))) float v8f;

// ---------------------------------------------------------------------------
// h = relu(x @ W0 + b0); also writes the GCNII residual x0 = h.
// One thread per (node, 4-feature chunk): x row loaded once, float4 stores.
// ---------------------------------------------------------------------------
__global__ void lin0_relu_kernel(const float* __restrict__ x,
                                 const float* __restrict__ W0,
                                 const float* __restrict__ b0,
                                 float* __restrict__ h,
                                 float* __restrict__ x0,
                                 int n_nodes) {
  int i = blockIdx.x * blockDim.x + threadIdx.x;
  int total = n_nodes * (HID / 4);
  if (i >= total) return;
  int node = i / (HID / 4);
  int f = (i - node * (HID / 4)) * 4;
  float xv0 = x[node * 3 + 0];
  float xv1 = x[node * 3 + 1];
  float xv2 = x[node * 3 + 2];
  float4 o;
  o.x = fmaxf(fmaf(xv2, W0[2 * HID + f + 0], fmaf(xv1, W0[HID + f + 0], fmaf(xv0, W0[f + 0], b0[f + 0]))), 0.0f);
  o.y = fmaxf(fmaf(xv2, W0[2 * HID + f + 1], fmaf(xv1, W0[HID + f + 1], fmaf(xv0, W0[f + 1], b0[f + 1]))), 0.0f);
  o.z = fmaxf(fmaf(xv2, W0[2 * HID + f + 2], fmaf(xv1, W0[HID + f + 2], fmaf(xv0, W0[f + 2], b0[f + 2]))), 0.0f);
  o.w = fmaxf(fmaf(xv2, W0[2 * HID + f + 3], fmaf(xv1, W0[HID + f + 3], fmaf(xv0, W0[f + 3], b0[f + 3]))), 0.0f);
  size_t off = (size_t)node * HID + f;
  *(float4*)(h + off) = o;
  *(float4*)(x0 + off) = o;
}

__global__ void zero4_kernel(float4* __restrict__ p, int n4) {
  int i = blockIdx.x * blockDim.x + threadIdx.x;
  if (i < n4) p[i] = make_float4(0.f, 0.f, 0.f, 0.f);
}

// ---------------------------------------------------------------------------
// agg[dst[e], :] += h[src[e], :]   (segment_sum via L2 float atomics)
// One thread per (edge, float4 chunk): global_load_b128 gather,
// 4x global_atomic_add_f32 scatter (no packed f32 atomic exists on CDNA5).
// ---------------------------------------------------------------------------
__global__ void scatter_add_kernel(const float* __restrict__ h,
                                   const int* __restrict__ src,
                                   const int* __restrict__ dst,
                                   float* __restrict__ agg,
                                   int n_edges) {
  int i = blockIdx.x * blockDim.x + threadIdx.x;
  int total = n_edges * (HID / 4);
  if (i >= total) return;
  int e = i / (HID / 4);
  int f = (i - e * (HID / 4)) * 4;
  int s = src[e];
  int d = dst[e];
  float4 v = *(const float4*)(h + (size_t)s * HID + f);
  float* dp = agg + (size_t)d * HID + f;
  atomicAdd(dp + 0, v.x);
  atomicAdd(dp + 1, v.y);
  atomicAdd(dp + 2, v.z);
  atomicAdd(dp + 3, v.w);
}

// ---------------------------------------------------------------------------
// WMMA GEMM tile kernel (one 16-row tile per wave32).
// mode == 0 (GCNII layer):
//   T = 0.9*A_in + 0.1*x0 ; out = relu((1-beta)*T + beta*(T @ W))
// mode == 1 (final linear):
//   T = A_in ; out = T @ W + bias
// V_WMMA_F32_16X16X4_F32, W held transposed in LDS so every A/B fragment is
// one contiguous 8-byte ds_load (pitch 50 -> conflict-free bank pairs).
// ---------------------------------------------------------------------------
__global__ __launch_bounds__(32 * WPB)
void gcn2_wmma_kernel(const float* __restrict__ A_in,
                      const float* __restrict__ x0,
                      const float* __restrict__ W,
                      const float* __restrict__ bias,
                      float* __restrict__ out,
                      float beta, int mode, int ntiles, int n_nodes) {
  __shared__ float ldsWt[HID * WP];        // Wt[col][k] = W[k][col], padded pitch
  __shared__ float ldsT[WPB][16 * TP];     // per-wave 16x48 activation tile, padded

  int wave = threadIdx.x >> 5;
  int lane = threadIdx.x & 31;
  int tile = blockIdx.x * WPB + wave;
  bool active = tile < ntiles;
  int m0 = tile * 16;

  // Stage W transposed into LDS (block-cooperative; coalesced global reads)
  for (int i = threadIdx.x; i < HID * HID; i += blockDim.x) {
    int k = i / HID;
    int col = i - k * HID;
    ldsWt[col * WP + k] = W[i];
  }

  // Stage this wave's T tile (16x48 = 768 elems, 24 per lane)
  if (active) {
    float* T = ldsT[wave];
#pragma unroll
    for (int j = 0; j < 24; ++j) {
      int idx = lane + 32 * j;
      int row = idx / HID;
      int col = idx - row * HID;
      int gr = m0 + row;
      float v = 0.0f;
      if (gr < n_nodes) {
        size_t g = (size_t)gr * HID + col;
        v = A_in[g];
        if (mode == 0) v = 0.9f * v + 0.1f * x0[g];
      }
      T[row * TP + col] = v;
    }
  }
  __syncthreads();
  if (!active) return;

  const float* T = ldsT[wave];
  int half = lane >> 4;     // 0: lanes 0-15 (K = k0,k0+1), 1: lanes 16-31 (K = k0+2,k0+3)
  int lr = lane & 15;

  v8f acc0 = {}, acc1 = {}, acc2 = {};
#pragma unroll
  for (int ks = 0; ks < KSTEPS; ++ks) {
    int k = 4 * ks + 2 * half;                       // even -> 8B-aligned v2f loads
    v2f a  = *(const v2f*)(T + lr * TP + k);         // A: row lr, K = k,k+1
    v2f bA = *(const v2f*)(ldsWt + (0 + lr) * WP + k);   // B: col lr,      K = k,k+1
    v2f bB = *(const v2f*)(ldsWt + (16 + lr) * WP + k);  // B: col 16+lr
    v2f bC = *(const v2f*)(ldsWt + (32 + lr) * WP + k);  // B: col 32+lr
    acc0 = __builtin_amdgcn_wmma_f32_16x16x4_f32(false, a, false, bA, (short)0, acc0, false, false);
    acc1 = __builtin_amdgcn_wmma_f32_16x16x4_f32(false, a, false, bB, (short)0, acc1, false, false);
    acc2 = __builtin_amdgcn_wmma_f32_16x16x4_f32(false, a, false, bC, (short)0, acc2, false, false);
  }

  // Epilogue. C/D layout: lane<16 -> (M=v, N=lane); lane>=16 -> (M=8+v, N=lane-16)
  float omb = 1.0f - beta;
#pragma unroll
  for (int v = 0; v < 8; ++v) {
    int M = v + 8 * half;
    int gr = m0 + M;
    if (gr >= n_nodes) continue;
#pragma unroll
    for (int nt = 0; nt < NTILES_N; ++nt) {
      int col = nt * 16 + lr;
      float accv = (nt == 0) ? acc0[v] : (nt == 1) ? acc1[v] : acc2[v];
      float o;
      if (mode == 0) {
        float tv = T[M * TP + col];
        o = fmaxf(omb * tv + beta * accv, 0.0f);
      } else {
        o = accv + bias[col];
      }
      out[(size_t)gr * HID + col] = o;
    }
  }
}

// ---------------------------------------------------------------------------
// In-place log_softmax over 48 columns; one wave32 per row.
// Lane i covers elements {i} and {i+32 if i<16}.
// ---------------------------------------------------------------------------
__global__ void log_softmax_kernel(float* __restrict__ out, int n_rows) {
  int warp = (blockIdx.x * blockDim.x + threadIdx.x) >> 5;
  int lane = threadIdx.x & 31;
  if (warp >= n_rows) return;
  float* p = out + (size_t)warp * HID;
  float a = p[lane];
  float b = (lane < 16) ? p[lane + 32] : -3.402823466e38f;
  float m = fmaxf(a, b);
#pragma unroll
  for (int off = 16; off > 0; off >>= 1) m = fmaxf(m, __shfl_xor(m, off, 32));
  float s = expf(a - m) + ((lane < 16) ? expf(b - m) : 0.0f);
#pragma unroll
  for (int off = 16; off > 0; off >>= 1) s += __shfl_xor(s, off, 32);
  float lse = m + logf(s);
  p[lane] = a - lse;
  if (lane < 16) p[lane + 32] = b - lse;
}

// ---------------------------------------------------------------------------
extern "C" void kernel_launch(void* const* d_in, const int* in_sizes, int n_in,
                              void* d_out, int out_size, void* d_ws, size_t ws_size,
                              hipStream_t stream) {
  const float* x     = (const float*)d_in[0];   // (N, 3)
  const int*   ei    = (const int*)d_in[1];     // (2, E)
  const float* W0    = (const float*)d_in[2];   // (3, 48)
  const float* b0    = (const float*)d_in[3];   // (48,)
  const float* convW = (const float*)d_in[4];   // (4, 48, 48)
  const float* W1    = (const float*)d_in[5];   // (48, 48)
  const float* b1    = (const float*)d_in[6];   // (48,)

  const int n_nodes = in_sizes[0] / 3;
  const int n_edges = in_sizes[1] / 2;
  const int* src = ei;
  const int* dst = ei + n_edges;

  float* fo  = (float*)d_out;
  float* ws  = (float*)d_ws;
  float* x0  = ws;                              // residual (N*48)
  float* h   = ws + (size_t)n_nodes * HID;      // activations (N*48)
  float* agg = fo;                              // reuse output buffer as agg scratch

  const int totNH4 = n_nodes * (HID / 4);       // float4 granularity
  const int totEF4 = n_edges * (HID / 4);
  dim3 blk(256);
  dim3 gNH4((totNH4 + 255) / 256);
  dim3 gEF4((totEF4 + 255) / 256);

  lin0_relu_kernel<<<gNH4, blk, 0, stream>>>(x, W0, b0, h, x0, n_nodes);

  const int ntiles = (n_nodes + 15) / 16;
  dim3 gemmGrid((ntiles + WPB - 1) / WPB);
  dim3 gemmBlk(32 * WPB);

  for (int l = 0; l < 4; ++l) {
    zero4_kernel<<<gNH4, blk, 0, stream>>>((float4*)agg, totNH4);
    scatter_add_kernel<<<gEF4, blk, 0, stream>>>(h, src, dst, agg, n_edges);
    float beta = logf(0.5f / (float)(l + 1) + 1.0f);
    gcn2_wmma_kernel<<<gemmGrid, gemmBlk, 0, stream>>>(
        agg, x0, convW + (size_t)l * HID * HID, nullptr, h, beta, 0, ntiles, n_nodes);
  }

  // Final linear into d_out (agg contents no longer needed)
  gcn2_wmma_kernel<<<gemmGrid, gemmBlk, 0, stream>>>(
      h, nullptr, W1, b1, fo, 0.0f, 1, ntiles, n_nodes);

  dim3 gSM((n_nodes + 7) / 8);  // 8 waves of 32 per 256-thread block
  log_softmax_kernel<<<gSM, blk, 0, stream>>>(fo, n_nodes);
}